// HdOVCModel_10050223473339
// MI455X (gfx1250) — compile-verified
//
#include <hip/hip_runtime.h>
#include <math.h>

// ---------------------------------------------------------------------------
// Problem constants (match reference)
// ---------------------------------------------------------------------------
#define P_TOT   10000
#define N_NEUR  512
#define L_BEAR  30

#define N_TILES 32            // N_NEUR / 16
#define P_CHUNKS 125
#define P_PER_BLOCK 80        // P_TOT / P_CHUNKS
#define WAVES_PER_BLOCK 8
#define P_PER_WAVE 10         // P_PER_BLOCK / WAVES_PER_BLOCK

typedef __attribute__((ext_vector_type(2))) float v2f;
typedef __attribute__((ext_vector_type(8))) float v8f;

__device__ __forceinline__ float period_bound_f(float d) {
    const float PIf = 3.14159265358979323846f;
    const float TWO_PI = 6.28318530717958647692f;
    float m = fmodf(d + PIf, TWO_PI);
    if (m < 0.0f) m += TWO_PI;       // jnp.mod: result has sign of divisor
    return m - PIf;
}

// ---------------------------------------------------------------------------
// Kernel 1: memory-bound bearing reduction via V_WMMA_F32_16X16X4_F32.
//   partial[chunk][n] = sum_{p in chunk} r_hpc[p] * dot(conn[p,n,:], r_bearing)
// Wave owns a 16-row n-tile. A operand: streamed conn rows, one 8-byte-aligned
// float2 per lane per K-chunk (exact ISA 16x4 f32 A layout). B operand:
// weights broadcast to all 16 columns, so every D column carries the dot.
// K-chunk 7 covers only l=28,29: hi lanes (K=30,31) read a harmless in-row
// dummy (l=26,27) whose B weight is 0 -> branch-free inner loop, no OOB.
// ---------------------------------------------------------------------------
__global__ __launch_bounds__(256)
void bearing_wmma_kernel(const float* __restrict__ conn,       // [P][N][L]
                         const float* __restrict__ r_hpc,      // [P]
                         const float* __restrict__ r_bearing,  // [L]
                         float* __restrict__ partial)          // [P_CHUNKS][N]
{
    const int tile  = blockIdx.x;       // 0..31  -> rows n0..n0+15
    const int chunk = blockIdx.y;       // 0..124 -> p range
    const int tid   = threadIdx.x;
    const int lane  = tid & 31;
    const int wave  = tid >> 5;
    const int laneLo = lane & 15;
    const bool hi   = (lane >= 16);
    const int n0    = tile * 16;

    // B-operand patterns (before the per-p r_hpc scale):
    // lane<16 holds K = 4q,4q+1 ; lane>=16 holds K = 4q+2,4q+3 ; l>=30 -> 0
    float rb0[8], rb1[8];
#pragma unroll
    for (int q = 0; q < 8; ++q) {
        int k0 = 4 * q + (hi ? 2 : 0);
        rb0[q] = (k0     < L_BEAR) ? r_bearing[k0]     : 0.0f;
        rb1[q] = (k0 + 1 < L_BEAR) ? r_bearing[k0 + 1] : 0.0f;
    }

    v8f acc = {0.f, 0.f, 0.f, 0.f, 0.f, 0.f, 0.f, 0.f};

    const int p_base = chunk * P_PER_BLOCK + wave * P_PER_WAVE;

    // Prefetch this wave's 10 contiguous r_hpc gains out of the hot loop.
    float hp[P_PER_WAVE];
#pragma unroll
    for (int i = 0; i < P_PER_WAVE; ++i) hp[i] = r_hpc[p_base + i];

    // Per-lane row bases (float offsets are always even -> 8B aligned):
    //   chunks 0..6 : rowA + 4q       (lo lanes K=4q,4q+1; hi lanes K=4q+2,4q+3)
    //   chunk  7    : row7            (lo lanes l=28,29;   hi lanes dummy l=26,27)
    const size_t rowOff = (size_t)(n0 + laneLo) * L_BEAR;
    const float* rowA = conn + rowOff + (hi ? 2 : 0);
    const float* row7 = conn + rowOff + (hi ? 26 : 28);

    for (int i = 0; i < P_PER_WAVE; ++i) {
        const size_t pOff = (size_t)(p_base + i) * (size_t)(N_NEUR * L_BEAR);
        const float* rp  = rowA + pOff;
        const float* rp7 = row7 + pOff;
        const float h = hp[i];
#pragma unroll
        for (int q = 0; q < 8; ++q) {
            v2f a = (q < 7) ? *(const v2f*)(rp + 4 * q)
                            : *(const v2f*)(rp7);
            v2f b;
            b.x = rb0[q] * h;
            b.y = rb1[q] * h;
            acc = __builtin_amdgcn_wmma_f32_16x16x4_f32(
                false, a, false, b, (short)0, acc, false, false);
        }
    }

    // D layout (32-bit C/D 16x16): VGPR v, lanes 0-15 -> M=v, lanes 16-31 -> M=v+8.
    // All columns equal (B broadcast): lane 0 / lane 16 carry rows 0-7 / 8-15.
    __shared__ float red[WAVES_PER_BLOCK * 16];
    if (lane == 0) {
#pragma unroll
        for (int i = 0; i < 8; ++i) red[wave * 16 + i] = acc[i];
    }
    if (lane == 16) {
#pragma unroll
        for (int i = 0; i < 8; ++i) red[wave * 16 + 8 + i] = acc[i];
    }
    __syncthreads();
    if (tid < 16) {
        float s = 0.0f;
        for (int w = 0; w < WAVES_PER_BLOCK; ++w) s += red[w * 16 + tid];
        partial[(size_t)chunk * N_NEUR + n0 + tid] = s;
    }
}

// ---------------------------------------------------------------------------
// Kernel 2: everything small, in one 512-thread block (thread n == neuron n).
// ---------------------------------------------------------------------------
__global__ __launch_bounds__(512)
void finalize_kernel(const float* __restrict__ r,
                     const float* __restrict__ u,
                     const float* __restrict__ v,
                     const float* __restrict__ r_hpc,
                     const float* __restrict__ pos_estimate,
                     const float* __restrict__ angular_velocity,
                     const float* __restrict__ sen2HD_stre,
                     const float* __restrict__ shared_t,
                     const float* __restrict__ partial,   // [P_CHUNKS][N]
                     float* __restrict__ out)
{
    const float PIf    = 3.14159265358979323846f;
    const float TWO_PI = 6.28318530717958647692f;
    const float GW     = PIf * 0.25f;                    // GAUSS_WIDTH
    const float J0     = 20.0f / (float)N_NEUR;
    const float kdiv   = 0.8f / (float)N_NEUR * 20.0f;   // 0.03125
    const float DT     = 0.1f;
    const float TAU    = 1.0f;
    const float TAU_E  = 1000.0f;

    __shared__ float s_a[N_NEUR];     // cos(x)*r  -> later r1
    __shared__ float s_b[N_NEUR];     // sin(x)*r
    __shared__ float s_c[N_NEUR];     // hpc partials -> later conn[]
    __shared__ float s_r[N_NEUR];
    __shared__ float s_scal[4];       // [0]=pos_est', [1]=S_hpc, [2]=sum_r1

    const int n = threadIdx.x;
    const float x = -PIf + (TWO_PI / (float)N_NEUR) * (float)n;
    const float rn = r[n];

    s_r[n] = rn;
    s_a[n] = cosf(x) * rn;
    s_b[n] = sinf(x) * rn;
    {
        float hp = 0.0f;
        for (int j = n; j < P_TOT; j += N_NEUR) hp += r_hpc[j];
        s_c[n] = hp;
    }
    __syncthreads();

    if (n == 0) {
        float zre = 0.0f, zim = 0.0f, S = 0.0f;
        for (int i = 0; i < N_NEUR; ++i) { zre += s_a[i]; zim += s_b[i]; S += s_c[i]; }
        const float center = atan2f(zim, zre);
        const float pe = pos_estimate[0];
        float pe1 = pe + (period_bound_f(center - pe) / TAU_E + angular_velocity[0]) * DT;
        s_scal[0] = period_bound_f(pe1);
        s_scal[1] = S;
    }
    __syncthreads();

    const float pos_e = s_scal[0];
    const float S_hpc = s_scal[1];

    // recurrent connectivity row (x0 = -pi): |x0 - x_n| = n * 2pi/N
    {
        const float d = period_bound_f(fabsf(-PIf - x));
        const float t = d / GW;
        s_c[n] = J0 * expf(-0.5f * t * t) / (TWO_PI * GW * GW);
    }
    __syncthreads();

    // Irec[n] = sum_m r[m] * conn[(n-m) mod N]   (FFT circular conv identity)
    float Irec = 0.0f;
    for (int m = 0; m < N_NEUR; ++m) {
        Irec += s_r[m] * s_c[(n - m) & (N_NEUR - 1)];
    }

    // bearing input: fixed-order chunk reduction (deterministic), /= sum(r_hpc)
    float bear = 0.0f;
    for (int c = 0; c < P_CHUNKS; ++c) bear += partial[(size_t)c * N_NEUR + n];
    bear /= S_hpc;

    // motion input
    float dd = period_bound_f(x - pos_e) / GW;
    const float Iext = 3.0f * expf(-0.25f * dd * dd);

    const float itot = Iext + Irec + bear * sen2HD_stre[0];

    // exact exponential ODE step; v1 never reaches the output
    const float dt_ode = shared_t[1] - shared_t[0];
    const float e1 = expf(-dt_ode / TAU);
    const float C1 = itot - v[n];
    const float u1 = C1 + (u[n] - C1) * e1;
    const float un = fmaxf(u1, 0.0f);
    const float r1 = un * un;

    s_a[n] = r1;
    __syncthreads();
    if (n == 0) {
        float s = 0.0f;
        for (int i = 0; i < N_NEUR; ++i) s += s_a[i];
        s_scal[2] = s;
    }
    __syncthreads();

    out[n] = r1 / (1.0f + kdiv * s_scal[2]);
}

// ---------------------------------------------------------------------------
extern "C" void kernel_launch(void* const* d_in, const int* in_sizes, int n_in,
                              void* d_out, int out_size, void* d_ws, size_t ws_size,
                              hipStream_t stream) {
    const float* conn        = (const float*)d_in[0];   // (P,N,L)
    const float* r           = (const float*)d_in[1];   // (N,)
    const float* u           = (const float*)d_in[2];   // (N,)
    const float* v           = (const float*)d_in[3];   // (N,)
    const float* r_hpc       = (const float*)d_in[4];   // (P,)
    const float* r_bearing   = (const float*)d_in[5];   // (L,)
    const float* pos_est     = (const float*)d_in[6];   // (1,)
    const float* ang_vel     = (const float*)d_in[7];   // (1,)
    /* d_in[8] = HD, unused (get_HD = 0) */
    const float* sen2        = (const float*)d_in[9];   // (1,)
    const float* shared_t    = (const float*)d_in[10];  // (2,)
    float* out               = (float*)d_out;

    float* partial = (float*)d_ws;                       // P_CHUNKS*N floats = 256 KB

    dim3 gridB(N_TILES, P_CHUNKS);
    bearing_wmma_kernel<<<gridB, 256, 0, stream>>>(conn, r_hpc, r_bearing, partial);

    finalize_kernel<<<1, N_NEUR, 0, stream>>>(r, u, v, r_hpc, pos_est, ang_vel,
                                              sen2, shared_t, partial, out);
}